// POS_62869731278966
// MI455X (gfx1250) — compile-verified
//
#include <hip/hip_runtime.h>
#include <hip/hip_bf16.h>
#include <complex>
#include <cmath>

#ifndef M_PI
#define M_PI 3.14159265358979323846
#endif

// ---------------- constants ----------------
#define BB 32        // batch
#define TT 300       // time
#define CC 3         // channels
#define HW 5184      // 72*72
#define WW 29        // window-1 samples per window (l-1)
#define TS 270       // T - l
#define NN 300       // detrend size
#define LAM2 10000.0 // LAMBDA^2

typedef __attribute__((ext_vector_type(2))) float v2f;
typedef __attribute__((ext_vector_type(8))) float v8f;

struct Filt { float b[6]; float a[6]; float zi[5]; };

// ---------------- kernel 1: spatial mean (the only HBM-bound stage) ----------------
__global__ void mean_kernel(const float* __restrict__ x, float* __restrict__ m) {
    __shared__ float s[256];
    int slice = blockIdx.x;                         // (b*T + t)*C + c
    const float4* p = (const float4*)(x + (size_t)slice * HW);
    float sum = 0.0f;
    for (int i = threadIdx.x; i < HW / 4; i += 256) {
        float4 v = p[i];
        sum += (v.x + v.y) + (v.z + v.w);
    }
    s[threadIdx.x] = sum;
    __syncthreads();
    for (int off = 128; off > 0; off >>= 1) {
        if (threadIdx.x < off) s[threadIdx.x] += s[threadIdx.x + off];
        __syncthreads();
    }
    if (threadIdx.x == 0) m[slice] = s[0] * (1.0f / (float)HW);
}

// ---------------- kernel 2: POS projection per sliding window ----------------
__global__ void pos_kernel(const float* __restrict__ m, float* __restrict__ Pn) {
    int tid = blockIdx.x * blockDim.x + threadIdx.x;
    if (tid >= BB * TS) return;
    int b = tid / TS, t = tid % TS;
    const float* M0 = m + ((size_t)b * TT + t) * CC;

    // per-channel temporal mean over the window
    float sr = 0, sg = 0, sbl = 0;
    for (int j = 0; j < WW; ++j) {
        sr += M0[j * 3 + 0]; sg += M0[j * 3 + 1]; sbl += M0[j * 3 + 2];
    }
    float ir = (float)WW / sr, ig = (float)WW / sg, ib = (float)WW / sbl;

    // accumulate moments of S0 = g-b, S1 = -2r+g+b
    float s0 = 0, s1 = 0, s00 = 0, s11 = 0, s01 = 0;
    for (int j = 0; j < WW; ++j) {
        float r = M0[j * 3 + 0] * ir, g = M0[j * 3 + 1] * ig, bl = M0[j * 3 + 2] * ib;
        float S0 = g - bl;
        float S1 = -2.0f * r + g + bl;
        s0 += S0; s1 += S1; s00 += S0 * S0; s11 += S1 * S1; s01 += S0 * S1;
    }
    const float inv = 1.0f / (float)WW;
    float m0 = s0 * inv, m1 = s1 * inv;
    float v0 = s00 * inv - m0 * m0;
    float v1 = s11 * inv - m1 * m1;
    float cv = s01 * inv - m0 * m1;
    float alpha = sqrtf(fmaxf(v0, 0.0f) / fmaxf(v1, 1e-30f));
    float meanP = m0 + alpha * m1;
    float varP = v0 + alpha * alpha * v1 + 2.0f * alpha * cv;
    float invStdP = rsqrtf(fmaxf(varP, 1e-30f));

    float* out = Pn + (size_t)tid * WW;
    for (int j = 0; j < WW; ++j) {
        float r = M0[j * 3 + 0] * ir, g = M0[j * 3 + 1] * ig, bl = M0[j * 3 + 2] * ib;
        float S0 = g - bl;
        float S1 = -2.0f * r + g + bl;
        float P = S0 + alpha * S1;
        out[j] = (P - meanP) * invStdP;
    }
}

// ---------------- kernel 3: overlap-add as deterministic gather ----------------
__global__ void overlap_kernel(const float* __restrict__ Pn, float* __restrict__ Hs) {
    int tid = blockIdx.x * blockDim.x + threadIdx.x;
    if (tid >= BB * TT) return;
    int b = tid / TT, i = tid % TT;
    int lo = i - (TS - 1); if (lo < 0) lo = 0;
    int hi = i < (WW - 1) ? i : (WW - 1);
    float s = 0.0f;
    for (int j = lo; j <= hi; ++j)
        s += Pn[((size_t)b * TS + (i - j)) * WW + j];
    Hs[tid] = s;
}

// ---------------- kernel 4a: build A = I + lam^2 D^T D, banded Cholesky, M = I - A^{-1} ----
__device__ inline double Dval(int k, int j) {
    if (k < 0 || k > NN - 3) return 0.0;
    if (j == k)     return 1.0;
    if (j == k + 1) return -2.0;
    if (j == k + 2) return 1.0;
    return 0.0;
}

__global__ void detrend_prepare(double* __restrict__ band,
                                double* __restrict__ solve,
                                float* __restrict__ Mmat) {
    double* a0 = band;          // A[i,i]
    double* a1 = band + NN;     // A[i+1,i]
    double* a2 = band + 2 * NN; // A[i+2,i]
    double* l0 = band + 3 * NN;
    double* l1 = band + 4 * NN;
    double* l2 = band + 5 * NN;
    int tid = threadIdx.x;

    for (int i = tid; i < NN; i += blockDim.x) {
        double s0 = 0, s1 = 0, s2 = 0;
        for (int k = i - 2; k <= i; ++k) {
            double di = Dval(k, i);
            s0 += di * di;
            s1 += di * Dval(k, i + 1);
            s2 += di * Dval(k, i + 2);
        }
        a0[i] = 1.0 + LAM2 * s0;
        a1[i] = LAM2 * s1;
        a2[i] = LAM2 * s2;
    }
    __syncthreads();

    if (tid == 0) {
        for (int i = 0; i < NN; ++i) {
            double s = a0[i];
            if (i >= 1) s -= l1[i - 1] * l1[i - 1];
            if (i >= 2) s -= l2[i - 2] * l2[i - 2];
            l0[i] = sqrt(s);
            if (i + 1 < NN) {
                double t = a1[i];
                if (i >= 1) t -= l1[i - 1] * l2[i - 1];
                l1[i] = t / l0[i];
            }
            if (i + 2 < NN) l2[i] = a2[i] / l0[i];
        }
    }
    __syncthreads();

    // one column of A^{-1} per thread: A y = e_j, then M[i,j] = delta - y[i]
    for (int j = tid; j < NN; j += blockDim.x) {
        double* y = solve + (size_t)j * NN;
        for (int i = 0; i < NN; ++i) {
            double s = (i == j) ? 1.0 : 0.0;
            if (i >= 1) s -= l1[i - 1] * y[i - 1];
            if (i >= 2) s -= l2[i - 2] * y[i - 2];
            y[i] = s / l0[i];
        }
        for (int i = NN - 1; i >= 0; --i) {
            double s = y[i];
            if (i + 1 < NN) s -= l1[i] * y[i + 1];
            if (i + 2 < NN) s -= l2[i] * y[i + 2];
            y[i] = s / l0[i];
        }
        for (int i = 0; i < NN; ++i)
            Mmat[(size_t)i * NN + j] = (float)(((i == j) ? 1.0 : 0.0) - y[i]);
    }
}

// ---------------- kernel 4b: d = Hs(32x300) * M(300x300) via V_WMMA_F32_16X16X4_F32 ----
// One wave per 16x16 output tile. A layout (16x4 f32): lane 0-15 row M=lane,
// VGPR0 holds K=0 (lanes 0-15) / K=2 (lanes 16-31), VGPR1 holds K=1 / K=3.
// B layout (4x16): same K split per half-wave, N = lane&15. D: VGPR r -> row r (+8 for hi half).
__global__ void detrend_gemm(const float* __restrict__ Hs,
                             const float* __restrict__ Mmat,
                             float* __restrict__ dbuf) {
    int lane = threadIdx.x;
    int half = lane >> 4;
    int l16  = lane & 15;
    int row0 = blockIdx.y * 16;
    int col0 = blockIdx.x * 16;

    int bCol = col0 + l16;
    float cmask = (bCol < NN) ? 1.0f : 0.0f;   // arithmetic mask, no EXEC divergence
    int bc = (bCol < NN) ? bCol : (NN - 1);

    int aRow = row0 + l16;                     // always < 32
    v8f acc = {0.0f, 0.0f, 0.0f, 0.0f, 0.0f, 0.0f, 0.0f, 0.0f};

    for (int k0 = 0; k0 < NN; k0 += 4) {
        int ka = k0 + (half << 1);
        v2f a, bvec;
        a.x = Hs[aRow * NN + ka];
        a.y = Hs[aRow * NN + ka + 1];
        bvec.x = Mmat[(size_t)ka * NN + bc] * cmask;
        bvec.y = Mmat[(size_t)(ka + 1) * NN + bc] * cmask;
        acc = __builtin_amdgcn_wmma_f32_16x16x4_f32(
            false, a, false, bvec, (short)0, acc, false, false);
    }

    if (bCol < NN) {
        int mbase = row0 + half * 8;
#pragma unroll
        for (int r = 0; r < 8; ++r)
            dbuf[(size_t)(mbase + r) * NN + bCol] = acc[r];
    }
}

// ---------------- kernel 5: filtfilt + min-max normalize; lane = batch row ----------------
__global__ void filtfilt_kernel(const float* __restrict__ dbuf,
                                float* __restrict__ ext,
                                float* __restrict__ ybuf,
                                float* __restrict__ out, Filt F) {
    const int PAD = 18, EL = TT + 2 * PAD; // 336
    int b = threadIdx.x;                   // 32 lanes = 32 batch rows
    const float* X = dbuf + (size_t)b * TT;
    float* E = ext + (size_t)b * EL;
    float* Y = ybuf + (size_t)b * EL;

    for (int i = 0; i < PAD; ++i) E[i] = 2.0f * X[0] - X[PAD - i];
    for (int j = 0; j < TT; ++j)  E[PAD + j] = X[j];
    for (int i = 0; i < PAD; ++i) E[TT + PAD + i] = 2.0f * X[TT - 1] - X[TT - 2 - i];

    float z[5];
#pragma unroll
    for (int n = 0; n < 5; ++n) z[n] = F.zi[n] * E[0];
    for (int t = 0; t < EL; ++t) {
        float xt = E[t];
        float y = F.b[0] * xt + z[0];
#pragma unroll
        for (int n = 0; n < 4; ++n) z[n] = z[n + 1] + F.b[n + 1] * xt - F.a[n + 1] * y;
        z[4] = F.b[5] * xt - F.a[5] * y;
        Y[t] = y;
    }
#pragma unroll
    for (int n = 0; n < 5; ++n) z[n] = F.zi[n] * Y[EL - 1];
    for (int t = 0; t < EL; ++t) {
        float xt = Y[EL - 1 - t];
        float y = F.b[0] * xt + z[0];
#pragma unroll
        for (int n = 0; n < 4; ++n) z[n] = z[n + 1] + F.b[n + 1] * xt - F.a[n + 1] * y;
        z[4] = F.b[5] * xt - F.a[5] * y;
        E[t] = y;
    }
    // final f[j] = E[EL-1-PAD-j] = E[317-j]
    float mn = 1e30f, mx = -1e30f;
    for (int j = 0; j < TT; ++j) {
        float v = E[EL - 1 - PAD - j];
        mn = fminf(mn, v); mx = fmaxf(mx, v);
    }
    float inv = 1.0f / (mx - mn);
    for (int j = 0; j < TT; ++j)
        out[(size_t)b * TT + j] = (E[EL - 1 - PAD - j] - mn) * inv;
}

// ---------------- host: Butterworth(5) + zi (input-independent, capture-safe) ----------------
static void compute_filter(Filt& F) {
    using cd = std::complex<double>;
    const int order = 5;
    cd p[5];
    for (int k = 1; k <= order; ++k)
        p[k - 1] = std::exp(cd(0.0, M_PI * (2.0 * k + order - 1.0) / (2.0 * order)));
    const double fs = 2.0;
    const double wn = 3.0 / 15.0;                    // CUTOFF / (FS/2)
    const double warped = 2.0 * fs * std::tan(M_PI * wn / fs);
    for (int i = 0; i < order; ++i) p[i] *= warped;
    const double kg = std::pow(warped, (double)order);
    const double fs2 = 2.0 * fs;
    cd pz[5]; cd prod = 1.0;
    for (int i = 0; i < order; ++i) { pz[i] = (fs2 + p[i]) / (fs2 - p[i]); prod *= (fs2 - p[i]); }
    const double kz = kg * (1.0 / prod).real();
    const double binom[6] = {1, 5, 10, 10, 5, 1};    // poly(-ones(5))
    double bc[6], ac[6];
    for (int i = 0; i < 6; ++i) bc[i] = kz * binom[i];
    cd acc[6] = {cd(1.0), cd(0.0), cd(0.0), cd(0.0), cd(0.0), cd(0.0)};
    for (int r = 0; r < order; ++r)
        for (int j = r + 1; j >= 1; --j) acc[j] -= pz[r] * acc[j - 1];
    for (int i = 0; i < 6; ++i) ac[i] = acc[i].real();

    // zi: solve (I - comp^T) zi = b[1:] - a[1:]*b[0]
    double A[5][6];
    for (int i = 0; i < 5; ++i) {
        for (int j = 0; j < 5; ++j) {
            double compT = (j == 0) ? -ac[i + 1] : ((i == j - 1) ? 1.0 : 0.0);
            A[i][j] = ((i == j) ? 1.0 : 0.0) - compT;
        }
        A[i][5] = bc[i + 1] - ac[i + 1] * bc[0];
    }
    for (int c = 0; c < 5; ++c) {           // Gaussian elimination, partial pivot
        int piv = c;
        for (int r = c + 1; r < 5; ++r)
            if (std::fabs(A[r][c]) > std::fabs(A[piv][c])) piv = r;
        for (int j = 0; j < 6; ++j) { double t = A[c][j]; A[c][j] = A[piv][j]; A[piv][j] = t; }
        for (int r = 0; r < 5; ++r) {
            if (r == c) continue;
            double f = A[r][c] / A[c][c];
            for (int j = c; j < 6; ++j) A[r][j] -= f * A[c][j];
        }
    }
    for (int i = 0; i < 6; ++i) { F.b[i] = (float)bc[i]; F.a[i] = (float)ac[i]; }
    for (int i = 0; i < 5; ++i) F.zi[i] = (float)(A[i][5] / A[i][i]);
}

extern "C" void kernel_launch(void* const* d_in, const int* in_sizes, int n_in,
                              void* d_out, int out_size, void* d_ws, size_t ws_size,
                              hipStream_t stream) {
    const float* x = (const float*)d_in[0];
    float* out = (float*)d_out;

    // workspace layout (doubles first for alignment)
    char* w = (char*)d_ws;
    double* band   = (double*)w;                        // 6*300 doubles
    double* solveb = band + 6 * NN;                     // 300*300 doubles
    float*  m      = (float*)(solveb + NN * NN);        // 28800
    float*  Pn     = m + BB * TT * CC;                  // 250560
    float*  Hs     = Pn + BB * TS * WW;                 // 9600
    float*  Mmat   = Hs + BB * TT;                      // 90000
    float*  dbuf   = Mmat + NN * NN;                    // 9600
    float*  ext    = dbuf + BB * TT;                    // 32*336
    float*  ybuf   = ext + BB * 336;                    // 32*336

    Filt F;
    compute_filter(F);

    mean_kernel<<<BB * TT * CC, 256, 0, stream>>>(x, m);
    pos_kernel<<<(BB * TS + 255) / 256, 256, 0, stream>>>(m, Pn);
    overlap_kernel<<<(BB * TT + 255) / 256, 256, 0, stream>>>(Pn, Hs);
    detrend_prepare<<<1, 320, 0, stream>>>(band, solveb, Mmat);
    detrend_gemm<<<dim3(19, 2, 1), 32, 0, stream>>>(Hs, Mmat, dbuf);
    filtfilt_kernel<<<1, 32, 0, stream>>>(dbuf, ext, ybuf, out, F);
}